// HebbianLayer_36137854829213
// MI455X (gfx1250) — compile-verified
//
#include <hip/hip_runtime.h>
#include <math.h>

#define IN_DIM   16384
#define OUT_DIM  4096
#define ALPHA    0.001f
#define GAMMA    0.99f
#define EPS      0.0001f
#define BROWS    64                 // rows per scan block
#define NBLK     (OUT_DIM / BROWS)  // 64 row blocks
#define IN4      (IN_DIM / 4)       // columns in float4 units

typedef __attribute__((ext_vector_type(2))) float v2f;
typedef __attribute__((ext_vector_type(4))) float v4f;
typedef __attribute__((ext_vector_type(8))) float v8f;

// ---------------------------------------------------------------------------
// Phase 1: h = relu(W @ x) via V_WMMA_F32_16X16X4_F32.
// Block r (256 thr = 8 waves) computes rows [16r, 16r+16). Each wave owns a
// K-slice of 2048. A-matrix = x-chunk broadcast over all 16 M rows, B-matrix =
// 16-row tile of W, so D[m][n] = dot(x_chunk, W[rb+n]) for every m.
// Per ISA layouts (CDNA5 ISA):
//   A 16x4 f32 : lanes 0-15 hold K=0,1 (v0,v1); lanes 16-31 hold K=2,3
//   B 4x16 f32 : v0 = rows K=0|K=1 (lane halves), v1 = rows K=2|K=3
//   D 16x16 f32: v0, lane n -> (M=0,N=n) / (M=8,N=n-16)  => c[0] has our dot
// ---------------------------------------------------------------------------
__global__ void __launch_bounds__(256)
hebb_gemv_wmma(const float* __restrict__ W, const float* __restrict__ x,
               float* __restrict__ h) {
  __shared__ float red[8 * 16];
  const int lane = threadIdx.x & 31;
  const int wave = threadIdx.x >> 5;
  const int rb   = blockIdx.x * 16;
  const int row  = rb + (lane & 15);
  const float* __restrict__ wrow = W + (size_t)row * IN_DIM;
  const int kbase = wave * (IN_DIM / 8);
  const bool lo   = lane < 16;

  v8f c = {};
  #pragma unroll 4
  for (int k = 0; k < IN_DIM / 8; k += 4) {
    const int kb = kbase + k;
    const v4f xv = *(const v4f*)(x + kb);            // one b128, half-uniform
    const v4f wv = *(const v4f*)(wrow + kb);         // one b128, full line use
    v2f a, b;
    a.x = lo ? xv.x : xv.z;                          // A: K=0,1 | K=2,3
    a.y = lo ? xv.y : xv.w;
    b.x = lo ? wv.x : wv.y;                          // B: rows K=0|1, K=2|3
    b.y = lo ? wv.z : wv.w;
    c = __builtin_amdgcn_wmma_f32_16x16x4_f32(false, a, false, b,
                                              (short)0, c, false, false);
  }
  // lane L holds the partial dot for row rb+(L&15) in c[0]
  if (lane < 16) red[wave * 16 + lane] = c[0];
  __syncthreads();
  if (threadIdx.x < 16) {
    float s = 0.f;
    #pragma unroll
    for (int w = 0; w < 8; ++w) s += red[w * 16 + threadIdx.x];
    h[rb + threadIdx.x] = fmaxf(s, 0.f);             // relu
  }
}

// ---------------------------------------------------------------------------
// Phase 2: all O(OUT_DIM) stats in one 1024-thread block (4 elems/thread).
// y = h^2/max^2 ; nea = g*ea + (1-g)*y ; g_pos = EPS*tanh(-EPS*(nea/avg-1))+1
// ---------------------------------------------------------------------------
__global__ void __launch_bounds__(1024)
hebb_stats(const float* __restrict__ h, const float* __restrict__ exp_avg,
           float* __restrict__ y_out, float* __restrict__ nea_out,
           float* __restrict__ y_ws, float* __restrict__ gp,
           float* __restrict__ gn) {
  __shared__ float red[1024];
  const int t = threadIdx.x;
  float hv[4];
  float m = 0.f;
  #pragma unroll
  for (int i = 0; i < 4; ++i) { hv[i] = h[t * 4 + i]; m = fmaxf(m, hv[i]); }
  red[t] = m;
  __syncthreads();
  for (int off = 512; off > 0; off >>= 1) {
    if (t < off) red[t] = fmaxf(red[t], red[t + off]);
    __syncthreads();
  }
  const float mx = red[0];
  __syncthreads();
  const float inv = 1.f / (mx * mx);
  float yv[4], nv[4], sum = 0.f;
  #pragma unroll
  for (int i = 0; i < 4; ++i) {
    yv[i] = hv[i] * hv[i] * inv;
    nv[i] = GAMMA * exp_avg[t * 4 + i] + (1.f - GAMMA) * yv[i];
    sum += nv[i];
    y_out[t * 4 + i]   = yv[i];
    y_ws[t * 4 + i]    = yv[i];
    nea_out[t * 4 + i] = nv[i];
  }
  red[t] = sum;
  __syncthreads();
  for (int off = 512; off > 0; off >>= 1) {
    if (t < off) red[t] += red[t + off];
    __syncthreads();
  }
  const float iavg = (float)OUT_DIM / red[0];
  #pragma unroll
  for (int i = 0; i < 4; ++i) {
    const float An = nv[i] * iavg;
    const float g  = EPS * tanhf(-EPS * (An - 1.f)) + 1.f;
    gp[t * 4 + i] = g;
    gn[t * 4 + i] = 1.f / g;
  }
}

// ---------------------------------------------------------------------------
// Phase 3a: block partial sums S[b][j] = sum_{l in block b} y[l] * W[l][j]
// float4 column group per thread: b128 loads, 4 FMA chains. W read stays RT
// (re-read by phase 3c; want it resident in the 192MB L2).
// ---------------------------------------------------------------------------
__global__ void __launch_bounds__(256)
hebb_partial(const float* __restrict__ W, const float* __restrict__ y,
             float* __restrict__ S) {
  __shared__ float ys[BROWS];
  const int b  = blockIdx.y;
  const int j4 = blockIdx.x * 256 + threadIdx.x;     // float4 column group
  if (threadIdx.x < BROWS) ys[threadIdx.x] = y[b * BROWS + threadIdx.x];
  __syncthreads();
  const v4f* __restrict__ wp = (const v4f*)(W + (size_t)b * BROWS * IN_DIM) + j4;
  v4f s = {0.f, 0.f, 0.f, 0.f};
  #pragma unroll 4
  for (int l = 0; l < BROWS; ++l) {
    const v4f w  = wp[(size_t)l * IN4];
    const float yl = ys[l];
    s.x = fmaf(yl, w.x, s.x);
    s.y = fmaf(yl, w.y, s.y);
    s.z = fmaf(yl, w.z, s.z);
    s.w = fmaf(yl, w.w, s.w);
  }
  ((v4f*)(S + (size_t)b * IN_DIM))[j4] = s;
}

// ---------------------------------------------------------------------------
// Phase 3b: exclusive scan over the 64 blocks, per column group, in place.
// ---------------------------------------------------------------------------
__global__ void __launch_bounds__(256)
hebb_scan(float* __restrict__ S) {
  const int j4 = blockIdx.x * 256 + threadIdx.x;
  v4f run = {0.f, 0.f, 0.f, 0.f};
  #pragma unroll 4
  for (int b = 0; b < NBLK; ++b) {
    v4f* p = (v4f*)S + (size_t)b * IN4 + j4;
    const v4f t = *p;
    *p = run;
    run.x += t.x; run.y += t.y; run.z += t.z; run.w += t.w;
  }
}

// ---------------------------------------------------------------------------
// Phase 3c: seeded inclusive scan + Hebbian update + decay, fused store of W2.
//   s (incl.) = prefix + sum_{k<=l} y[k]W[k,j];  A = y[l]*s
//   W1 = W + ALPHA*(y[l]*x[j] - A);  W2 = W1>0 ? W1*gp : (W1<0 ? W1*gn : 0)
// W load is the LAST use -> nontemporal load (don't keep in cache).
// W2 store is write-once -> nontemporal store (don't evict W from L2).
// ---------------------------------------------------------------------------
__global__ void __launch_bounds__(256)
hebb_update(const float* __restrict__ W, const float* __restrict__ x,
            const float* __restrict__ y, const float* __restrict__ gp,
            const float* __restrict__ gn, const float* __restrict__ S,
            float* __restrict__ Wout) {
  __shared__ float ys[BROWS], gps[BROWS], gns[BROWS];
  const int b  = blockIdx.y;
  const int j4 = blockIdx.x * 256 + threadIdx.x;
  if (threadIdx.x < BROWS) {
    ys[threadIdx.x]  = y[b * BROWS + threadIdx.x];
    gps[threadIdx.x] = gp[b * BROWS + threadIdx.x];
    gns[threadIdx.x] = gn[b * BROWS + threadIdx.x];
  }
  __syncthreads();
  v4f s = ((const v4f*)(S + (size_t)b * IN_DIM))[j4];
  const v4f xj = ((const v4f*)x)[j4];
  const v4f* __restrict__ wp = (const v4f*)(W    + (size_t)b * BROWS * IN_DIM) + j4;
  v4f* __restrict__       op = (v4f*)      (Wout + (size_t)b * BROWS * IN_DIM) + j4;
  #pragma unroll 4
  for (int l = 0; l < BROWS; ++l) {
    const v4f w  = __builtin_nontemporal_load(wp + (size_t)l * IN4);
    const float yl  = ys[l];
    const float gpl = gps[l];
    const float gnl = gns[l];
    v4f w2;
    {
      s.x = fmaf(yl, w.x, s.x);
      const float w1 = w.x + ALPHA * (yl * xj.x - yl * s.x);
      w2.x = (w1 > 0.f) ? w1 * gpl : ((w1 < 0.f) ? w1 * gnl : 0.f);
    }
    {
      s.y = fmaf(yl, w.y, s.y);
      const float w1 = w.y + ALPHA * (yl * xj.y - yl * s.y);
      w2.y = (w1 > 0.f) ? w1 * gpl : ((w1 < 0.f) ? w1 * gnl : 0.f);
    }
    {
      s.z = fmaf(yl, w.z, s.z);
      const float w1 = w.z + ALPHA * (yl * xj.z - yl * s.z);
      w2.z = (w1 > 0.f) ? w1 * gpl : ((w1 < 0.f) ? w1 * gnl : 0.f);
    }
    {
      s.w = fmaf(yl, w.w, s.w);
      const float w1 = w.w + ALPHA * (yl * xj.w - yl * s.w);
      w2.w = (w1 > 0.f) ? w1 * gpl : ((w1 < 0.f) ? w1 * gnl : 0.f);
    }
    __builtin_nontemporal_store(w2, op + (size_t)l * IN4);
  }
}

// ---------------------------------------------------------------------------
extern "C" void kernel_launch(void* const* d_in, const int* in_sizes, int n_in,
                              void* d_out, int out_size, void* d_ws, size_t ws_size,
                              hipStream_t stream) {
  (void)in_sizes; (void)n_in; (void)out_size; (void)ws_size;
  const float* x  = (const float*)d_in[0];   // (16384,)
  const float* W  = (const float*)d_in[1];   // (4096, 16384)
  const float* ea = (const float*)d_in[2];   // (4096,)

  float* out   = (float*)d_out;
  float* y_out = out;                                      // (4096,)
  float* W_out = out + OUT_DIM;                            // (4096*16384,)
  float* n_out = out + OUT_DIM + (size_t)OUT_DIM * IN_DIM; // (4096,)

  float* ws = (float*)d_ws;
  float* h    = ws;                 // 4096
  float* y_ws = ws + OUT_DIM;       // 4096
  float* gp   = ws + 2 * OUT_DIM;   // 4096
  float* gn   = ws + 3 * OUT_DIM;   // 4096
  float* S    = ws + 4 * OUT_DIM;   // 64 * 16384 (~4 MB)

  // Phase 1: WMMA GEMV, 256 blocks x 8 waves
  hebb_gemv_wmma<<<OUT_DIM / 16, 256, 0, stream>>>(W, x, h);
  // Phase 2: stats
  hebb_stats<<<1, 1024, 0, stream>>>(h, ea, y_out, n_out, y_ws, gp, gn);
  // Phase 3: decoupled column scan + fused update (float4 column groups)
  dim3 g3(IN4 / 256, NBLK);
  hebb_partial<<<g3, 256, 0, stream>>>(W, y_ws, S);
  hebb_scan<<<IN4 / 256, 256, 0, stream>>>(S);
  hebb_update<<<g3, 256, 0, stream>>>(W, x, y_ws, gp, gn, S, W_out);
}